// MessagePassing_64888365908156
// MI455X (gfx1250) — compile-verified
//
#include <hip/hip_runtime.h>

// ---------------------------------------------------------------------------
// GAT (3 layers) on MI455X / gfx1250.
// GEMM uses V_WMMA_F32_16X16X4_F32 (exact fp32). Edge softmax/scatter phases
// are L2-bandwidth bound (h and accum each 25.6MB -> L2-resident on 192MB L2).
// ---------------------------------------------------------------------------

#define N_NODES 50000
#define N_EDGES 800000
#define TOT_E   (N_EDGES + N_NODES)   // with self loops
#define F       128                    // feature dim in AND out (H*C = 128)
#define NHEAD   4
#define CH      32
#define SLOPE   0.2f

typedef float v2f __attribute__((ext_vector_type(2)));
typedef float v8f __attribute__((ext_vector_type(8)));

// ---------------- dense GEMM: Hout[n, o] = sum_k X[n,k] * W[o,k] -----------
// grid.x = N/16 row tiles; 256 threads = 8 waves; wave w -> output cols [16w,16w+16)
__global__ __launch_bounds__(256) void gat_gemm_wmma(const float* __restrict__ X,
                                                     const float* __restrict__ W,
                                                     float* __restrict__ Hout) {
  __shared__ float lds_x[16 * F];             // 8 KB A-tile shared by 8 waves
  const int row0 = blockIdx.x * 16;
  const int tid  = threadIdx.x;
  {
    const float4* src4 = (const float4*)(X + (size_t)row0 * F);
    float4* dst4 = (float4*)lds_x;            // 2048 floats = 512 float4
    dst4[tid]       = src4[tid];
    dst4[tid + 256] = src4[tid + 256];
  }
  __syncthreads();

  const int wave = tid >> 5;
  const int lane = tid & 31;
  const int col0 = wave * 16;
  const int mn   = lane & 15;                  // A-row / B-col / D-col index
  const int khi  = (lane >> 4) * 2;            // lanes 0-15 hold K 0,1 ; 16-31 hold K 2,3

  const float* xrow = lds_x + mn * F;          // A fragment source (LDS)
  const float* wrow = W + (size_t)(col0 + mn) * F; // B fragment source (L2-resident)

  v8f acc = {};
  #pragma unroll
  for (int k = 0; k < F; k += 4) {
    v2f a, b;
    a.x = xrow[k + khi];
    a.y = xrow[k + khi + 1];
    b.x = wrow[k + khi];
    b.y = wrow[k + khi + 1];
    acc = __builtin_amdgcn_wmma_f32_16x16x4_f32(false, a, false, b,
                                                (short)0, acc, false, false);
  }
  // D layout: lanes 0-15 rows vr, lanes 16-31 rows vr+8 ; col = lane&15
  const int mbase = row0 + ((lane >> 4) << 3);
  #pragma unroll
  for (int vr = 0; vr < 8; ++vr)
    Hout[(size_t)(mbase + vr) * F + col0 + mn] = acc[vr];
}

// ---------------- per-node attention logits --------------------------------
__global__ void gat_attn(const float* __restrict__ Hf,
                         const float* __restrict__ att_s,
                         const float* __restrict__ att_d,
                         float* __restrict__ asrc, float* __restrict__ adst) {
  int idx = blockIdx.x * blockDim.x + threadIdx.x;  // node*NHEAD + head
  if (idx >= N_NODES * NHEAD) return;
  const int hd   = idx & 3;
  const int node = idx >> 2;
  const float* hv = Hf + (size_t)node * F + hd * CH;
  const float* as = att_s + hd * CH;
  const float* ad = att_d + hd * CH;
  float s = 0.f, d = 0.f;
  #pragma unroll
  for (int c = 0; c < CH; ++c) { s += hv[c] * as[c]; d += hv[c] * ad[c]; }
  asrc[idx] = s;
  adst[idx] = d;
}

// ---------------- per-layer state reset ------------------------------------
__global__ void gat_init(float* __restrict__ accum, unsigned* __restrict__ amax_u,
                         float* __restrict__ denom) {
  int idx = blockIdx.x * blockDim.x + threadIdx.x;
  if (idx < N_NODES * F) accum[idx] = 0.f;
  if (idx < N_NODES * NHEAD) { amax_u[idx] = 0u; denom[idx] = 0.f; }
}

// monotonic float<->uint order-preserving encoding (for atomicMax)
__device__ __forceinline__ unsigned enc_f(float f) {
  unsigned b = __float_as_uint(f);
  return (b & 0x80000000u) ? ~b : (b | 0x80000000u);
}
__device__ __forceinline__ float dec_f(unsigned u) {
  unsigned b = (u & 0x80000000u) ? (u & 0x7fffffffu) : ~u;
  return __uint_as_float(b);
}
__device__ __forceinline__ float lrelu(float a) { return a > 0.f ? a : SLOPE * a; }

// ---------------- pass 1: segment max of edge logits -----------------------
__global__ void gat_edge_max(const int* __restrict__ src, const int* __restrict__ dst,
                             const float* __restrict__ asrc, const float* __restrict__ adst,
                             unsigned* __restrict__ amax_u) {
  int idx = blockIdx.x * blockDim.x + threadIdx.x;  // edge*NHEAD + head
  if (idx >= TOT_E * NHEAD) return;
  const int e  = idx >> 2;
  const int hd = idx & 3;
  const int s  = (e < N_EDGES) ? src[e] : (e - N_EDGES);
  const int d  = (e < N_EDGES) ? dst[e] : (e - N_EDGES);
  const float a = lrelu(asrc[s * NHEAD + hd] + adst[d * NHEAD + hd]);
  atomicMax(&amax_u[d * NHEAD + hd], enc_f(a));
}

// ---------------- pass 2: exp + segment sum --------------------------------
__global__ void gat_edge_exp(const int* __restrict__ src, const int* __restrict__ dst,
                             const float* __restrict__ asrc, const float* __restrict__ adst,
                             const unsigned* __restrict__ amax_u,
                             float* __restrict__ denom, float* __restrict__ exbuf) {
  int idx = blockIdx.x * blockDim.x + threadIdx.x;
  if (idx >= TOT_E * NHEAD) return;
  const int e  = idx >> 2;
  const int hd = idx & 3;
  const int s  = (e < N_EDGES) ? src[e] : (e - N_EDGES);
  const int d  = (e < N_EDGES) ? dst[e] : (e - N_EDGES);
  const float a  = lrelu(asrc[s * NHEAD + hd] + adst[d * NHEAD + hd]);
  const float mx = dec_f(amax_u[d * NHEAD + hd]);
  const float ex = __expf(a - mx);
  exbuf[idx] = ex;
  unsafeAtomicAdd(&denom[d * NHEAD + hd], ex);
}

// ---------------- pass 3: weighted message scatter-add ---------------------
// one wave32 per edge; lane l handles channel l of each head (coalesced).
__global__ __launch_bounds__(256) void gat_edge_scatter(
    const int* __restrict__ src, const int* __restrict__ dst,
    const float* __restrict__ Hf, const float* __restrict__ exbuf,
    const float* __restrict__ denom, float* __restrict__ accum) {
  int gid  = blockIdx.x * blockDim.x + threadIdx.x;
  int e    = gid >> 5;
  int lane = gid & 31;
  if (e >= TOT_E) return;
  const int s = (e < N_EDGES) ? src[e] : (e - N_EDGES);
  const int d = (e < N_EDGES) ? dst[e] : (e - N_EDGES);
  const float4 ex4 = *(const float4*)(exbuf + (size_t)e * 4);
  const float4 dn4 = *(const float4*)(denom + (size_t)d * 4);
  const float c0 = ex4.x / fmaxf(dn4.x, 1e-16f);
  const float c1 = ex4.y / fmaxf(dn4.y, 1e-16f);
  const float c2 = ex4.z / fmaxf(dn4.z, 1e-16f);
  const float c3 = ex4.w / fmaxf(dn4.w, 1e-16f);
  const float* hv = Hf + (size_t)s * F;
  float* ov       = accum + (size_t)d * F;
  unsafeAtomicAdd(&ov[lane],      hv[lane]      * c0);
  unsafeAtomicAdd(&ov[32 + lane], hv[32 + lane] * c1);
  unsafeAtomicAdd(&ov[64 + lane], hv[64 + lane] * c2);
  unsafeAtomicAdd(&ov[96 + lane], hv[96 + lane] * c3);
}

// ---------------- finalize: +bias, ReLU ------------------------------------
__global__ void gat_finalize(const float* __restrict__ accum, const float* __restrict__ bias,
                             float* __restrict__ y) {
  int idx = blockIdx.x * blockDim.x + threadIdx.x;
  if (idx >= N_NODES * F) return;
  float v = accum[idx] + bias[idx & (F - 1)];
  y[idx] = fmaxf(v, 0.f);
}

extern "C" void kernel_launch(void* const* d_in, const int* in_sizes, int n_in,
                              void* d_out, int out_size, void* d_ws, size_t ws_size,
                              hipStream_t stream) {
  (void)in_sizes; (void)n_in; (void)out_size; (void)ws_size;

  const float* x   = (const float*)d_in[0];
  const int*   ei  = (const int*)d_in[1];
  const int*   src = ei;                 // edge_index[0]
  const int*   dst = ei + N_EDGES;       // edge_index[1]
  const float* Wl[3]  = {(const float*)d_in[2], (const float*)d_in[6],  (const float*)d_in[10]};
  const float* asl[3] = {(const float*)d_in[3], (const float*)d_in[7],  (const float*)d_in[11]};
  const float* adl[3] = {(const float*)d_in[4], (const float*)d_in[8],  (const float*)d_in[12]};
  const float* bl[3]  = {(const float*)d_in[5], (const float*)d_in[9],  (const float*)d_in[13]};

  char* ws = (char*)d_ws;
  float*    hbuf   = (float*)ws;    ws += (size_t)N_NODES * F * sizeof(float);
  float*    ybuf   = (float*)ws;    ws += (size_t)N_NODES * F * sizeof(float);
  float*    accum  = (float*)ws;    ws += (size_t)N_NODES * F * sizeof(float);
  float*    asrc   = (float*)ws;    ws += (size_t)N_NODES * NHEAD * sizeof(float);
  float*    adst_  = (float*)ws;    ws += (size_t)N_NODES * NHEAD * sizeof(float);
  float*    denom  = (float*)ws;    ws += (size_t)N_NODES * NHEAD * sizeof(float);
  unsigned* amax_u = (unsigned*)ws; ws += (size_t)N_NODES * NHEAD * sizeof(unsigned);
  float*    exbuf  = (float*)ws;    ws += (size_t)TOT_E * NHEAD * sizeof(float);

  const float* fin = x;
  for (int l = 0; l < 3; ++l) {
    float* yout = (l == 2) ? (float*)d_out : ybuf;
    gat_gemm_wmma<<<N_NODES / 16, 256, 0, stream>>>(fin, Wl[l], hbuf);
    gat_attn<<<(N_NODES * NHEAD + 255) / 256, 256, 0, stream>>>(hbuf, asl[l], adl[l], asrc, adst_);
    gat_init<<<(N_NODES * F + 255) / 256, 256, 0, stream>>>(accum, amax_u, denom);
    gat_edge_max<<<(TOT_E * NHEAD + 255) / 256, 256, 0, stream>>>(src, dst, asrc, adst_, amax_u);
    gat_edge_exp<<<(TOT_E * NHEAD + 255) / 256, 256, 0, stream>>>(src, dst, asrc, adst_, amax_u, denom, exbuf);
    gat_edge_scatter<<<((size_t)TOT_E * 32 + 255) / 256, 256, 0, stream>>>(src, dst, hbuf, exbuf, denom, accum);
    gat_finalize<<<(N_NODES * F + 255) / 256, 256, 0, stream>>>(accum, bl[l], yout);
    fin = ybuf;
  }
}